// TMANorm_85031762526554
// MI455X (gfx1250) — compile-verified
//
#include <hip/hip_runtime.h>
#include <hip/hip_bf16.h>

typedef __attribute__((ext_vector_type(2))) float v2f;
typedef __attribute__((ext_vector_type(8))) float v8f;
typedef __attribute__((__vector_size__(4 * sizeof(int)))) int v4i;

#define T_LEN   16384
#define NPER    128
#define STEP    64
#define NSEG    255
#define NF      65
#define NB      32
#define NC      64
#define CH_TOT  2048
#define HALF_T  8192

#if defined(__AMDGCN__) && __has_builtin(__builtin_amdgcn_global_load_async_to_lds_b128) && __has_builtin(__builtin_amdgcn_s_wait_asynccnt)
#define USE_ASYNC 1
#else
#define USE_ASYNC 0
#endif

// ---------------------------------------------------------------------------
// Stage 1: Welch PSD via rfft-as-GEMM with V_WMMA_F32_16X16X4_F32.
// One block per channel, 9 waves; wave w owns 16 basis columns (interleaved
// cos/sin per frequency bin), cached in registers; loops 16 segment tiles.
// ---------------------------------------------------------------------------
__global__ __launch_bounds__(288) void psd_kernel(const float* __restrict__ x,
                                                  float* __restrict__ psd) {
  const int ch   = blockIdx.x;
  const int lane = threadIdx.x & 31;
  const int wv   = threadIdx.x >> 5;        // 0..8 : column tile
  __shared__ float sAcc[160];
  for (int i = threadIdx.x; i < 160; i += 288) sAcc[i] = 0.0f;
  __syncthreads();

  const int n     = lane & 15;              // B column within tile / D column
  const int c     = (wv << 4) + n;          // 0..143 (cols 130..143 are pad)
  const int f     = c >> 1;
  const int isSin = c & 1;
  const int khalf = (lane & 16) ? 2 : 0;    // K sub-offset per half-wave
  const float w0  = 6.2831853071795864769f / 128.0f;

  // Basis fragments (K = 128 -> 32 k-steps), resident in VGPRs.
  v2f breg[32];
  for (int kk = 0; kk < 32; ++kk) {
    int k0 = (kk << 2) + khalf;
    v2f b; b.x = 0.0f; b.y = 0.0f;
    if (f < NF) {
      float t0 = w0 * (float)(k0 * f);
      float t1 = w0 * (float)((k0 + 1) * f);
      b.x = isSin ? __sinf(t0) : __cosf(t0);
      b.y = isSin ? __sinf(t1) : __cosf(t1);
    }
    breg[kk] = b;
  }

  const long chbase = (long)ch * T_LEN;
  const long LIM    = (long)CH_TOT * T_LEN - 126;  // keep v2f loads in bounds
  float part = 0.0f;

  for (int st = 0; st < 16; ++st) {
    const int s = (st << 4) + n;            // segment row (A-matrix row)
    long rowbase = chbase + (long)s * STEP + khalf;
    if (rowbase > LIM) rowbase = LIM;       // only triggers for pad row s==255
    v8f acc = {};
#pragma unroll
    for (int kk = 0; kk < 32; ++kk) {
      v2f a = *(const v2f*)(x + rowbase + (kk << 2));   // 8B-aligned
      acc = __builtin_amdgcn_wmma_f32_16x16x4_f32(
          false, a, false, breg[kk], (short)0, acc, false, false);
    }
    // accumulate |.|^2 per column; rows of D: v + (hi?8:0) within tile
    const int rbase = (st << 4) + ((lane & 16) ? 8 : 0);
#pragma unroll
    for (int v = 0; v < 8; ++v)
      if (rbase + v < NSEG) part += acc[v] * acc[v];
  }
  atomicAdd(&sAcc[c], part);                // ds_add_f32
  __syncthreads();

  if (threadIdx.x < NF) {
    const int fo = threadIdx.x;
    float s2 = sAcc[2 * fo] + sAcc[2 * fo + 1];         // Re^2 + Im^2 summed
    float factor = (fo == 0 || fo == NF - 1) ? 1.0f : 2.0f;
    psd[ch * NF + fo] = s2 * factor * (1.0f / (128.0f * 255.0f));
  }
}

// ---------------------------------------------------------------------------
// Stage 2: barycenter -> D -> irfft(128) -> fftshift -> flip  => H[2048][128]
// One block per C-channel (64 blocks), 128 threads.  H[i] = h[(63-i) & 127].
// ---------------------------------------------------------------------------
__global__ __launch_bounds__(128) void filt_kernel(const float* __restrict__ psd,
                                                   float* __restrict__ H) {
  const int c = blockIdx.x;                 // 0..63
  __shared__ float sq[NB][NF];
  __shared__ float sb[NF];
  __shared__ float ctab[128];
  const int tid = threadIdx.x;

  for (int i = tid; i < NB * NF; i += 128) {
    int b = i / NF, f = i % NF;
    sq[b][f] = sqrtf(psd[(b * NC + c) * NF + f]);
  }
  ctab[tid] = __cosf(6.2831853071795864769f / 128.0f * (float)tid);
  __syncthreads();

  if (tid < NF) {
    float s = 0.0f;
    for (int b = 0; b < NB; ++b) s += sq[b][tid];
    sb[tid] = s * (1.0f / (float)NF);       // sqrt(barycenter)
  }
  __syncthreads();

  for (int e = tid; e < NB * 128; e += 128) {
    const int b = e >> 7, i = e & 127;
    const int nn = (63 - i) & 127;          // fftshift + flip
    float acc = sb[0] / sq[b][0];
    for (int f = 1; f < 64; ++f) {
      float D = sb[f] / sq[b][f];
      acc += 2.0f * D * ctab[(f * nn) & 127];
    }
    acc += (sb[64] / sq[b][64]) * ((nn & 1) ? -1.0f : 1.0f);
    H[((b * NC + c) << 7) + i] = acc * (1.0f / 128.0f);
  }
}

// ---------------------------------------------------------------------------
// Stage 3: depthwise 128-tap conv as implicit GEMM (Toeplitz filter x signal).
// x window (left pad = 64 elems so global/LDS offsets are 16B-congruent) is
// filled with GLOBAL_LOAD_ASYNC_TO_LDS_B128 (ASYNCcnt) when available.
// Reduction index u = m+k+1: A[m][u] = H[u-m-1], B[u][n] = xs[q0+16n+u].
// 36 x V_WMMA_F32_16X16X4_F32 per 256-output tile; b128 stores.
// ---------------------------------------------------------------------------
__global__ __launch_bounds__(256) void conv_kernel(const float* __restrict__ x,
                                                   const float* __restrict__ H,
                                                   float* __restrict__ out) {
  const int blk  = blockIdx.x;              // 0..4095
  const int ch   = blk >> 1;
  const int half = blk & 1;
  const int base = half * HALF_T;
  const int lane = threadIdx.x & 31;
  const int wv   = threadIdx.x >> 5;        // 0..7

  __shared__ __align__(16) float xs[HALF_T + 192]; // xs[j] = x[base-64+j]
  __shared__ v2f aT[36 * 32];                      // Toeplitz filter fragments

  const long chbase = (long)ch * T_LEN;

#if USE_ASYNC
  // zero the out-of-signal pad region
  if (half == 0) {
    if (threadIdx.x < 64) xs[threadIdx.x] = 0.0f;
  } else {
    for (int j = 8256 + threadIdx.x; j < HALF_T + 144; j += 256) xs[j] = 0.0f;
  }
  {
    const int  copyStart = half ? 0 : 64;             // local element index
    const int  copyLen   = half ? 8256 : 8272;        // elements (mult of 16)
    const float* gsrc    = x + chbase + (half ? (HALF_T - 64) : 0);
    for (int j4 = threadIdx.x * 4; j4 < copyLen; j4 += 256 * 4) {
      __builtin_amdgcn_global_load_async_to_lds_b128(
          (v4i*)(gsrc + j4),
          (v4i*)(xs + copyStart + j4), 0, 0);
    }
  }
  __builtin_amdgcn_s_wait_asynccnt(0);
#else
  for (int j = threadIdx.x; j < HALF_T + 144; j += 256) {
    int g = base - 64 + j;
    xs[j] = (g >= 0 && g < T_LEN) ? x[chbase + g] : 0.0f;
  }
#endif

  const float* Hc = H + ((long)ch << 7);
  for (int e = threadIdx.x; e < 36 * 32; e += 256) {
    const int l = e & 31, kk = e >> 5;
    const int m  = l & 15;
    const int jb = (kk << 2) + ((l & 16) ? 2 : 0);
    const int d0 = jb - 1 - m, d1 = jb - m;          // H[u - m - 1]
    v2f v;
    v.x = (d0 >= 0 && d0 < 128) ? Hc[d0] : 0.0f;
    v.y = (d1 >= 0 && d1 < 128) ? Hc[d1] : 0.0f;
    aT[e] = v;
  }
  __syncthreads();

  const int n  = lane & 15;
  const int hi = (lane & 16) ? 1 : 0;

  for (int tile = wv; tile < 32; tile += 8) {
    const int q0 = tile << 8;
    const float* bp = xs + q0 + (n << 4) + (hi << 1);
    v8f acc = {};
#pragma unroll
    for (int kk = 0; kk < 36; ++kk) {
      v2f a = aT[(kk << 5) + lane];                 // ds_load_b64
      v2f b = *(const v2f*)(bp + (kk << 2));        // ds_load_b64 (8B aligned)
      acc = __builtin_amdgcn_wmma_f32_16x16x4_f32(
          false, a, false, b, (short)0, acc, false, false);
    }
    const int p0 = base + q0 + (n << 4) + (hi << 3);  // contiguous 8 floats
    float* op = out + chbase + p0;
    *(float4*)(op)     = make_float4(acc[0], acc[1], acc[2], acc[3]);
    *(float4*)(op + 4) = make_float4(acc[4], acc[5], acc[6], acc[7]);
  }
}

// ---------------------------------------------------------------------------
extern "C" void kernel_launch(void* const* d_in, const int* in_sizes, int n_in,
                              void* d_out, int out_size, void* d_ws, size_t ws_size,
                              hipStream_t stream) {
  const float* x = (const float*)d_in[0];
  float* out = (float*)d_out;

  float* psd = (float*)d_ws;                                    // 2048*65 f32
  float* H   = (float*)((char*)d_ws + (size_t)CH_TOT * NF * 4); // 2048*128 f32

  psd_kernel <<<CH_TOT, 288, 0, stream>>>(x, psd);
  filt_kernel<<<NC,     128, 0, stream>>>(psd, H);
  conv_kernel<<<CH_TOT * 2, 256, 0, stream>>>(x, H, out);
}